// GraphSAGE_51324859187411
// MI455X (gfx1250) — compile-verified
//
#include <hip/hip_runtime.h>
#include <math.h>

// ---------------------------------------------------------------------------
// GraphSAGE on MI455X (gfx1250). fp32 WMMA (V_WMMA_F32_16X16X4_F32) for all
// GEMMs (workload is L2-resident + atomic-bound, so fp32 tensor ops give
// reference-exact precision at zero roofline cost). A-tiles staged in LDS
// (320 KB/WGP) to kill the 8x-19x redundant global reads across waves.
// ---------------------------------------------------------------------------

typedef __attribute__((ext_vector_type(2))) float v2f;
typedef __attribute__((ext_vector_type(8))) float v8f;

#define N_NODES 50000
#define N_EDGES 800000
#define F_DIM   128
#define NBITS   64
#define NCLS    100
#define HD      300
#define LDS_STRIDE 132  // 128 + 4 pad: rows land on distinct banks (4m mod 64)

__device__ __forceinline__ v8f wmma_f32(v2f a, v2f b, v8f c) {
  // (neg_a, A, neg_b, B, c_mod, C, reuse_a, reuse_b)
  return __builtin_amdgcn_wmma_f32_16x16x4_f32(false, a, false, b, (short)0, c,
                                               false, false);
}

// --- degree: deg[dst] += 1 -------------------------------------------------
__global__ void __launch_bounds__(256)
degree_kernel(const int* __restrict__ dst, float* __restrict__ deg) {
  int e = blockIdx.x * blockDim.x + threadIdx.x;
  if (e < N_EDGES) atomicAdd(&deg[dst[e]], 1.0f);
}

// --- scatter: agg[dst] += x[src]; one wave (32 lanes) covers 128 channels --
__global__ void __launch_bounds__(256)
scatter_add_kernel(const float* __restrict__ x, const int* __restrict__ src,
                   const int* __restrict__ dst, float* __restrict__ agg) {
  int gid = blockIdx.x * blockDim.x + threadIdx.x;
  int e   = gid >> 5;
  if (e >= N_EDGES) return;
  int ch = (gid & 31) * 4;
  int s = src[e], d = dst[e];
  const float4 v = *reinterpret_cast<const float4*>(x + (size_t)s * F_DIM + ch);
  float* p = agg + (size_t)d * F_DIM + ch;
  atomicAdd(p + 0, v.x);
  atomicAdd(p + 1, v.y);
  atomicAdd(p + 2, v.z);
  atomicAdd(p + 3, v.w);
}

// --- fused SAGE layer: h = relu((agg/deg)@wl + bl + x@wr) ------------------
// One block (8 waves) per 16-row tile; waves cover the 8 column tiles of 128.
// Mean and root A-tiles staged once in LDS; WMMA A-fragments fed from LDS.
__global__ void __launch_bounds__(256)
sage_layer_kernel(const float* __restrict__ x, const float* __restrict__ agg,
                  const float* __restrict__ deg, const float* __restrict__ wl,
                  const float* __restrict__ bl, const float* __restrict__ wr,
                  float* __restrict__ h) {
  __shared__ float sMean[16 * LDS_STRIDE];
  __shared__ float sX[16 * LDS_STRIDE];

  const int tm = blockIdx.x;  // 3125 row tiles

  // cooperative fill: 16 rows x 128 cols per matrix, float4 granularity
  for (int i = threadIdx.x; i < 16 * 32; i += 256) {
    const int r    = i >> 5;
    const int c4   = (i & 31) * 4;
    const int grow = tm * 16 + r;
    const float invd = 1.0f / fmaxf(deg[grow], 1.0f);
    const float4 va = *reinterpret_cast<const float4*>(agg + (size_t)grow * F_DIM + c4);
    const float4 vx = *reinterpret_cast<const float4*>(x   + (size_t)grow * F_DIM + c4);
    float* pm = sMean + r * LDS_STRIDE + c4;
    pm[0] = va.x * invd; pm[1] = va.y * invd; pm[2] = va.z * invd; pm[3] = va.w * invd;
    float* px = sX + r * LDS_STRIDE + c4;
    px[0] = vx.x; px[1] = vx.y; px[2] = vx.z; px[3] = vx.w;
  }
  __syncthreads();

  const int lane = threadIdx.x & 31;
  const int tn   = threadIdx.x >> 5;       // column tile 0..7
  const int mloc = lane & 15;
  const int ncol = tn * 16 + mloc;
  const int kb   = (lane >= 16) ? 2 : 0;   // K sub-pair per ISA A layout
  const float* sm = sMean + mloc * LDS_STRIDE;
  const float* sx = sX    + mloc * LDS_STRIDE;

  v8f acc = {};
#pragma unroll 4
  for (int kk = 0; kk < F_DIM; kk += 4) {
    v2f a1, b1, a2, b2;
    a1.x = sm[kk + kb];
    a1.y = sm[kk + kb + 1];
    b1.x = wl[(kk + kb)     * F_DIM + ncol];
    b1.y = wl[(kk + kb + 1) * F_DIM + ncol];
    acc = wmma_f32(a1, b1, acc);
    a2.x = sx[kk + kb];
    a2.y = sx[kk + kb + 1];
    b2.x = wr[(kk + kb)     * F_DIM + ncol];
    b2.y = wr[(kk + kb + 1) * F_DIM + ncol];
    acc = wmma_f32(a2, b2, acc);
  }
  const float bias  = bl[ncol];
  const int   mbase = tm * 16 + ((lane >= 16) ? 8 : 0);
#pragma unroll
  for (int i = 0; i < 8; ++i) {
    float v = fmaxf(acc[i] + bias, 0.0f);
    h[(size_t)(mbase + i) * F_DIM + ncol] = v;
  }
}

// --- generic WMMA GEMM: C[M,N] = act(A[M,K] @ B[K,N] + bias) ---------------
// grid.y = row tile, grid.x * waves = column tiles (clamped). K in {64,128}.
// A row tile staged in LDS, shared by all waves of the block.
template <int ACT>
__global__ void __launch_bounds__(256)
wmma_gemm_bias_kernel(const float* __restrict__ A, const float* __restrict__ B,
                      const float* __restrict__ bias, float* __restrict__ C,
                      int N, int K) {
  __shared__ float sA[16 * LDS_STRIDE];

  const int tm = blockIdx.y;
  const int k4 = K >> 2;
  for (int i = threadIdx.x; i < 16 * k4; i += blockDim.x) {
    const int r  = i / k4;
    const int c4 = (i - r * k4) * 4;
    const float4 v = *reinterpret_cast<const float4*>(A + (size_t)(tm * 16 + r) * K + c4);
    float* p = sA + r * LDS_STRIDE + c4;
    p[0] = v.x; p[1] = v.y; p[2] = v.z; p[3] = v.w;
  }
  __syncthreads();

  const int lane   = threadIdx.x & 31;
  const int wave   = threadIdx.x >> 5;
  const int tilesN = (N + 15) >> 4;
  int tn = blockIdx.x * (blockDim.x >> 5) + wave;
  if (tn >= tilesN) tn = tilesN - 1;  // clamp: keep EXEC all-ones for WMMA
  const int mloc = lane & 15;
  const int ncol = tn * 16 + mloc;
  const bool nok = (ncol < N);
  const int kb = (lane >= 16) ? 2 : 0;
  const float* sa = sA + mloc * LDS_STRIDE;

  v8f acc = {};
  for (int kk = 0; kk < K; kk += 4) {
    v2f a, b;
    a.x = sa[kk + kb];
    a.y = sa[kk + kb + 1];
    b.x = nok ? B[(size_t)(kk + kb) * N + ncol]     : 0.0f;
    b.y = nok ? B[(size_t)(kk + kb + 1) * N + ncol] : 0.0f;
    acc = wmma_f32(a, b, acc);
  }
  if (nok) {
    const float bv    = bias[ncol];
    const int   mbase = tm * 16 + ((lane >= 16) ? 8 : 0);
#pragma unroll
    for (int i = 0; i < 8; ++i) {
      float v = acc[i] + bv;
      if (ACT == 1) v = fmaxf(v, 0.0f);
      C[(size_t)(mbase + i) * N + ncol] = v;
    }
  }
}

// --- WMMA GEMM with fused BatchNorm(eval) epilogue -------------------------
// pre = gamma*(A@B + bconv - rm)*rsqrt(rv+eps) + beta
__global__ void __launch_bounds__(256)
wmma_gemm_bn_kernel(const float* __restrict__ A, const float* __restrict__ B,
                    const float* __restrict__ bconv, const float* __restrict__ gamma,
                    const float* __restrict__ beta, const float* __restrict__ rm,
                    const float* __restrict__ rv, float* __restrict__ C,
                    int N, int K) {
  __shared__ float sA[16 * LDS_STRIDE];

  const int tm = blockIdx.y;
  const int k4 = K >> 2;
  for (int i = threadIdx.x; i < 16 * k4; i += blockDim.x) {
    const int r  = i / k4;
    const int c4 = (i - r * k4) * 4;
    const float4 v = *reinterpret_cast<const float4*>(A + (size_t)(tm * 16 + r) * K + c4);
    float* p = sA + r * LDS_STRIDE + c4;
    p[0] = v.x; p[1] = v.y; p[2] = v.z; p[3] = v.w;
  }
  __syncthreads();

  const int lane   = threadIdx.x & 31;
  const int wave   = threadIdx.x >> 5;
  const int tilesN = (N + 15) >> 4;
  int tn = blockIdx.x * (blockDim.x >> 5) + wave;
  if (tn >= tilesN) tn = tilesN - 1;
  const int mloc = lane & 15;
  const int ncol = tn * 16 + mloc;
  const bool nok = (ncol < N);
  const int kb = (lane >= 16) ? 2 : 0;
  const float* sa = sA + mloc * LDS_STRIDE;

  v8f acc = {};
  for (int kk = 0; kk < K; kk += 4) {
    v2f a, b;
    a.x = sa[kk + kb];
    a.y = sa[kk + kb + 1];
    b.x = nok ? B[(size_t)(kk + kb) * N + ncol]     : 0.0f;
    b.y = nok ? B[(size_t)(kk + kb + 1) * N + ncol] : 0.0f;
    acc = wmma_f32(a, b, acc);
  }
  if (nok) {
    const float s     = gamma[ncol] * rsqrtf(rv[ncol] + 1e-5f);
    const float shift = (bconv[ncol] - rm[ncol]) * s + beta[ncol];
    const int   mbase = tm * 16 + ((lane >= 16) ? 8 : 0);
#pragma unroll
    for (int i = 0; i < 8; ++i)
      C[(size_t)(mbase + i) * N + ncol] = acc[i] * s + shift;
  }
}

// --- elementwise tanh ------------------------------------------------------
__global__ void __launch_bounds__(256)
tanh_kernel(const float* __restrict__ in, float* __restrict__ out, int n) {
  int i = blockIdx.x * blockDim.x + threadIdx.x;
  if (i < n) out[i] = tanhf(in[i]);
}

// --- true_lab: [N,64] @ [64,1] + btl ---------------------------------------
__global__ void __launch_bounds__(256)
true_lab_kernel(const float* __restrict__ pre, const float* __restrict__ wtl,
                const float* __restrict__ btl, float* __restrict__ out) {
  int r = blockIdx.x * blockDim.x + threadIdx.x;
  if (r >= N_NODES) return;
  const float* p = pre + (size_t)r * NBITS;
  float s = 0.0f;
#pragma unroll
  for (int k = 0; k < NBITS; ++k) s += p[k] * wtl[k];
  out[r] = s + btl[0];
}

// ---------------------------------------------------------------------------
extern "C" void kernel_launch(void* const* d_in, const int* in_sizes, int n_in,
                              void* d_out, int out_size, void* d_ws, size_t ws_size,
                              hipStream_t stream) {
  (void)in_sizes; (void)n_in; (void)out_size; (void)ws_size;

  const float* features = (const float*)d_in[0];
  const int*   edges    = (const int*)d_in[1];
  const float* w1l   = (const float*)d_in[2];
  const float* b1l   = (const float*)d_in[3];
  const float* w1r   = (const float*)d_in[4];
  const float* w2l   = (const float*)d_in[5];
  const float* b2l   = (const float*)d_in[6];
  const float* w2r   = (const float*)d_in[7];
  const float* whd   = (const float*)d_in[8];
  const float* bhd   = (const float*)d_in[9];
  const float* wclas = (const float*)d_in[10];
  const float* bclas = (const float*)d_in[11];
  const float* wconv = (const float*)d_in[12];
  const float* bconv = (const float*)d_in[13];
  const float* gamma = (const float*)d_in[14];
  const float* beta  = (const float*)d_in[15];
  const float* rm    = (const float*)d_in[16];
  const float* rv    = (const float*)d_in[17];
  const float* wtl   = (const float*)d_in[18];
  const float* btl   = (const float*)d_in[19];
  const float* wcv   = (const float*)d_in[20];
  const float* bcv   = (const float*)d_in[21];

  const int* src = edges;
  const int* dst = edges + N_EDGES;

  // workspace layout (floats): deg | agg | h1 | h2 | pre_bn  (~90 MB)
  float* ws     = (float*)d_ws;
  float* deg    = ws;
  float* agg    = deg + N_NODES;
  float* h1     = agg + (size_t)N_NODES * F_DIM;
  float* h2     = h1  + (size_t)N_NODES * F_DIM;
  float* pre_bn = h2  + (size_t)N_NODES * F_DIM;

  // output layout: logists | out(tanh) | fea_lab | fea_convert | true_lab
  float* out       = (float*)d_out;
  float* o_logists = out;
  float* o_tanh    = o_logists + (size_t)N_NODES * NCLS;
  float* o_fealab  = o_tanh    + (size_t)N_NODES * NBITS;
  float* o_feacv   = o_fealab  + (size_t)N_NODES * HD;
  float* o_truelab = o_feacv   + (size_t)N_NODES * NCLS;

  const int scatterBlocks = (N_EDGES * 32 + 255) / 256;
  const int rowTiles      = N_NODES / 16;  // 3125, exact
  auto gridFor = [rowTiles](int N, int waves) {
    int tilesN = (N + 15) / 16;
    return dim3((tilesN + waves - 1) / waves, rowTiles);
  };

  // ---- layer 1 ----
  hipMemsetAsync(deg, 0, sizeof(float) * ((size_t)N_NODES * (F_DIM + 1)), stream);
  degree_kernel<<<(N_EDGES + 255) / 256, 256, 0, stream>>>(dst, deg);
  scatter_add_kernel<<<scatterBlocks, 256, 0, stream>>>(features, src, dst, agg);
  sage_layer_kernel<<<rowTiles, 256, 0, stream>>>(features, agg, deg, w1l, b1l, w1r, h1);

  // ---- layer 2 ----
  hipMemsetAsync(agg, 0, sizeof(float) * (size_t)N_NODES * F_DIM, stream);
  scatter_add_kernel<<<scatterBlocks, 256, 0, stream>>>(h1, src, dst, agg);
  sage_layer_kernel<<<rowTiles, 256, 0, stream>>>(h1, agg, deg, w2l, b2l, w2r, h2);

  // ---- heads ----
  wmma_gemm_bias_kernel<0><<<gridFor(HD, 8), 256, 0, stream>>>(
      h2, whd, bhd, o_fealab, HD, F_DIM);
  wmma_gemm_bias_kernel<0><<<gridFor(NCLS, 8), 256, 0, stream>>>(
      h2, wclas, bclas, o_logists, NCLS, F_DIM);
  wmma_gemm_bn_kernel<<<gridFor(NBITS, 4), 128, 0, stream>>>(
      h2, wconv, bconv, gamma, beta, rm, rv, pre_bn, NBITS, F_DIM);
  tanh_kernel<<<(N_NODES * NBITS + 255) / 256, 256, 0, stream>>>(
      pre_bn, o_tanh, N_NODES * NBITS);
  wmma_gemm_bias_kernel<0><<<gridFor(NCLS, 8), 256, 0, stream>>>(
      o_tanh, wcv, bcv, o_feacv, NCLS, NBITS);
  true_lab_kernel<<<(N_NODES + 255) / 256, 256, 0, stream>>>(
      pre_bn, wtl, btl, o_truelab);
}